// CustomLoss_67989332295833
// MI455X (gfx1250) — compile-verified
//
#include <hip/hip_runtime.h>

namespace {
constexpr int kB   = 8192;   // rows
constexpr int kN   = 4096;   // cols
constexpr int kTPB = 256;    // 8 waves (wave32)
constexpr float kEps = 1e-12f;
}

typedef __attribute__((ext_vector_type(2))) float v2f;
typedef __attribute__((ext_vector_type(8))) float v8f;

// One block per row. Stream x with b128 loads, stage p=sigmoid(x) in LDS,
// then compute h + (l0+l1)/N from LDS (neighbors come for free), reduce.
__global__ __launch_bounds__(kTPB)
void row_loss_kernel(const float* __restrict__ x,
                     const float* __restrict__ t,
                     float* __restrict__ row_loss) {
  __shared__ float p_s[kN];
  __shared__ float wred[kTPB / 32];

  const int row = blockIdx.x;
  const int tid = threadIdx.x;
  const float* __restrict__ xr = x + (size_t)row * kN;

  // Pass 1: p = sigmoid(x), staged to LDS (aligned float4 in/out).
#pragma unroll
  for (int k = 0; k < kN / (kTPB * 4); ++k) {
    const int base = (k * kTPB + tid) * 4;
    if (k + 1 < kN / (kTPB * 4))
      __builtin_prefetch(xr + base + kTPB * 4, 0, 0);  // global_prefetch_b8
    const float4 v = *reinterpret_cast<const float4*>(xr + base);
    float4 p;
    p.x = 1.0f / (1.0f + __expf(-v.x));
    p.y = 1.0f / (1.0f + __expf(-v.y));
    p.z = 1.0f / (1.0f + __expf(-v.z));
    p.w = 1.0f / (1.0f + __expf(-v.w));
    *reinterpret_cast<float4*>(p_s + base) = p;
  }
  __syncthreads();

  // Pass 2: per-thread partial of  Σ p*neigh*logp + (Σ logp + Σ log(1-p))/N
  float acc = 0.0f;
#pragma unroll
  for (int k = 0; k < kN / (kTPB * 4); ++k) {
    const int base = (k * kTPB + tid) * 4;
    const float4 p = *reinterpret_cast<const float4*>(p_s + base);
    const float pl = (base > 0)        ? p_s[base - 1] : 0.0f;  // left halo
    const float pr = (base + 4 < kN)   ? p_s[base + 4] : 0.0f;  // right halo

    const float n0 = pl  + p.y;
    const float n1 = p.x + p.z;
    const float n2 = p.y + p.w;
    const float n3 = p.z + pr;

    const float lp0 = __logf(fmaxf(p.x, kEps));
    const float lp1 = __logf(fmaxf(p.y, kEps));
    const float lp2 = __logf(fmaxf(p.z, kEps));
    const float lp3 = __logf(fmaxf(p.w, kEps));
    const float lf0 = __logf(fmaxf(1.0f - p.x, kEps));
    const float lf1 = __logf(fmaxf(1.0f - p.y, kEps));
    const float lf2 = __logf(fmaxf(1.0f - p.z, kEps));
    const float lf3 = __logf(fmaxf(1.0f - p.w, kEps));

    const float h = p.x * n0 * lp0 + p.y * n1 * lp1 +
                    p.z * n2 * lp2 + p.w * n3 * lp3;
    const float l = (lp0 + lp1 + lp2 + lp3) + (lf0 + lf1 + lf2 + lf3);
    acc += h + l * (1.0f / kN);
  }

  // Deterministic block reduction: wave32 shuffle, then serial over 8 waves.
#pragma unroll
  for (int off = 16; off > 0; off >>= 1)
    acc += __shfl_down(acc, off, 32);
  if ((tid & 31) == 0) wred[tid >> 5] = acc;
  __syncthreads();
  if (tid == 0) {
    float tot = 0.0f;
#pragma unroll
    for (int w = 0; w < kTPB / 32; ++w) tot += wred[w];
    row_loss[row] = -t[row] * tot;   // loss_row = -t*(Σh + (Σlp+Σlf)/N)
  }
}

// Final 8192 -> 1 reduction on one wave using fp32 WMMA (16x16x4, f32 C tile).
// A = 16x4 chunk of row losses, B = ones => D[m][n] += Σ_k A[m,k].
// A layout (32-bit 16x4): lanes 0-15 hold (M=lane, K=0..1), lanes 16-31 (M, K=2..3).
__global__ __launch_bounds__(32)
void final_reduce_wmma(const float* __restrict__ rows, float* __restrict__ out) {
  const int lane = threadIdx.x;
  const int m  = lane & 15;
  const int kb = (lane >> 4) << 1;           // 0 for lanes 0-15, 2 for 16-31

  v8f acc = {0.f, 0.f, 0.f, 0.f, 0.f, 0.f, 0.f, 0.f};
  v2f bones; bones.x = 1.0f; bones.y = 1.0f; // B = all-ones 4x16

  for (int c = 0; c < kB / 64; ++c) {        // 128 chunks of 64 values
    const float* chunk = rows + c * 64 + m * 4 + kb;
    v2f a; a.x = chunk[0]; a.y = chunk[1];
    acc = __builtin_amdgcn_wmma_f32_16x16x4_f32(
        /*neg_a=*/false, a, /*neg_b=*/false, bones,
        /*c_mod=*/(short)0, acc, /*reuse_a=*/false, /*reuse_b=*/false);
  }

  // Columns are identical; lanes 0-15 hold M=0..7, lanes 16-31 hold M=8..15.
  float s = acc[0] + acc[1] + acc[2] + acc[3] +
            acc[4] + acc[5] + acc[6] + acc[7];
  s += __shfl_xor(s, 16, 32);
  if (lane == 0) out[0] = s * (1.0f / kB);   // mean over B
}

extern "C" void kernel_launch(void* const* d_in, const int* in_sizes, int n_in,
                              void* d_out, int out_size, void* d_ws, size_t ws_size,
                              hipStream_t stream) {
  (void)in_sizes; (void)n_in; (void)out_size; (void)ws_size;
  const float* x = (const float*)d_in[0];   // [8192, 4096] logits
  const float* t = (const float*)d_in[1];   // [8192] targets
  float* rows = (float*)d_ws;               // 8192 per-row losses (32 KB scratch)
  float* out  = (float*)d_out;              // scalar

  row_loss_kernel<<<kB, kTPB, 0, stream>>>(x, t, rows);
  final_reduce_wmma<<<1, 32, 0, stream>>>(rows, out);
}